// SphericalBasisLayer_67559835566340
// MI455X (gfx1250) — compile-verified
//
#include <hip/hip_runtime.h>
#include <hip/hip_bf16.h>

// SphericalBasisLayer (DimeNet-style) for MI455X / gfx1250.
// No contraction dim anywhere -> WMMA inapplicable; op is gather+elementwise.
// Design: (A) 42 norms -> ws, (B) rbf_env[M,42] table -> ws (84MB, L2-resident),
// (C) cbf_T[7,K] Legendre table -> ws (hoists cos+recurrence out of the 84M-elem
// kernel), (D) float2 gather*cbf with NON-TEMPORAL stores so the 336MB output
// stream does not evict the gather table from the 192MB L2.

#define NUM_S 7
#define NUM_R 6
#define SR    42
#define SR2   21          // float2 pairs per row
#define INV_CUTOFF 0.2f   // 1/5.0

typedef float v2f __attribute__((ext_vector_type(2)));   // native vector: OK for NT builtins

// Zeros of spherical Bessel j_n, n = 0..6, first 6 zeros each.
__constant__ float c_zeros[NUM_S * NUM_R] = {
    3.14159265f,  6.28318531f,  9.42477796f, 12.56637061f, 15.70796327f, 18.84955592f,
    4.49340946f,  7.72525184f, 10.90412166f, 14.06619391f, 17.22075527f, 20.37130296f,
    5.76345920f,  9.09501133f, 12.32294097f, 15.51460301f, 18.68903636f, 21.85387450f,
    6.98793200f, 10.41711855f, 13.69802315f, 16.92362129f, 20.12180617f, 23.30424717f,
    8.18256145f, 11.70490715f, 15.03966471f, 18.30125596f, 21.52541763f, 24.72756581f,
    9.35581211f, 12.96653017f, 16.35470964f, 19.65315286f, 22.90455056f, 26.12775495f,
   10.51283540f, 14.20739224f, 17.64797496f, 20.98346482f, 24.26276807f, 27.50786833f,
};

// sqrt((2l+1)/(4*pi)), l = 0..6
__constant__ float c_coef[NUM_S] = {
    0.28209479f, 0.48860251f, 0.63078313f, 0.74635267f,
    0.84628438f, 0.93560258f, 1.01782714f
};

// spherical Bessel j_n(x), upward recurrence, branch-free order select.
template <int MAXN>
__device__ __forceinline__ float sph_jn(int n, float x) {
    float sx = __sinf(x);
    float cx = __cosf(x);
    float inv = __builtin_amdgcn_rcpf(x);      // single v_rcp_f32
    float j0 = sx * inv;
    float res = j0;
    if (MAXN >= 1) {
        float j1 = (sx * inv - cx) * inv;
        res = (n >= 1) ? j1 : res;
        float jm1 = j0, jc = j1;
#pragma unroll
        for (int m = 2; m <= MAXN; ++m) {
            float jn = (float)(2 * m - 1) * inv * jc - jm1;
            jm1 = jc; jc = jn;
            res = (n == m) ? jn : res;
        }
    }
    return res;
}

// envelope(u), p = 6: 1/u - 28 u^5 + 48 u^6 - 21 u^7, 0 for u >= 1
__device__ __forceinline__ float envelope(float u) {
    float u2 = u * u;
    float u5 = u2 * u2 * u;
    float env = __builtin_amdgcn_rcpf(u) + u5 * (-28.0f + u * (48.0f + u * (-21.0f)));
    return (u < 1.0f) ? env : 0.0f;
}

// --- A: norm[s*6+r] = sqrt(2) / |j_{s+1}(z_{s,r})| ---
__global__ void sbl_init_norm(float* __restrict__ norm_out) {
    int t = threadIdx.x;
    if (t < SR) {
        int s = t / NUM_R;
        norm_out[t] = 1.41421356f / fabsf(sph_jn<NUM_S>(s + 1, c_zeros[t]));
    }
}

// --- B: rbf_env table [M, 42], float2 per thread, coalesced RT stores ---
__global__ void sbl_rbf_kernel(const float* __restrict__ dist,
                               const float* __restrict__ normt,
                               float* __restrict__ rbf,
                               unsigned int total /* = M*21 */) {
    unsigned int t = blockIdx.x * blockDim.x + threadIdx.x;
    if (t >= total) return;
    unsigned int m  = t / SR2;
    unsigned int j  = (t - m * SR2) * 2u;      // even element index in row
    unsigned int s0 = j / NUM_R;
    unsigned int s1 = (j + 1u) / NUM_R;

    float d = dist[m];                          // 21x reuse -> L0 hit
    float u = d * INV_CUTOFF;
    float env = envelope(u);                    // shared by the pair

    v2f v;
    v.x = env * normt[j]      * sph_jn<NUM_S - 1>((int)s0, u * c_zeros[j]);
    v.y = env * normt[j + 1u] * sph_jn<NUM_S - 1>((int)s1, u * c_zeros[j + 1u]);
    *(v2f*)(rbf + (size_t)m * SR + j) = v;      // 8B-aligned (42m+j even)
}

// --- C: cbf_T[7, K] = coef[l] * P_l(cos(theta_k)); coalesced stores per l ---
__global__ void sbl_cbf_kernel(const float* __restrict__ angles,
                               float* __restrict__ cbfT,
                               unsigned int K) {
    unsigned int k = blockIdx.x * blockDim.x + threadIdx.x;
    if (k >= K) return;
    float x = __cosf(angles[k]);
    float pm1 = 1.0f, pc = x;
    cbfT[k]     = c_coef[0] * pm1;
    cbfT[K + k] = c_coef[1] * pc;
#pragma unroll
    for (int l = 2; l < NUM_S; ++l) {
        float pl = ((float)(2 * l - 1) * x * pc - (float)(l - 1) * pm1) * (1.0f / (float)l);
        pm1 = pc; pc = pl;
        cbfT[(unsigned)l * K + k] = c_coef[l] * pl;
    }
}

// --- D: gather * cbf, float2 per thread, NT stores for the 336MB stream ---
__global__ void sbl_out_kernel(const long long* __restrict__ aidx,  // row 1 at +K
                               const float* __restrict__ rbf,
                               const float* __restrict__ cbfT,
                               float* __restrict__ out,
                               unsigned int K, unsigned int total /* = K*21 */) {
    unsigned int t = blockIdx.x * blockDim.x + threadIdx.x;
    if (t >= total) return;
    unsigned int k  = t / SR2;
    unsigned int j  = (t - k * SR2) * 2u;
    unsigned int s0 = j / NUM_R;
    unsigned int s1 = (j + 1u) / NUM_R;

    long long idx = aidx[(size_t)K + k];        // 21x reuse -> L0 hit
    const float* row = rbf + (size_t)idx * SR;
    __builtin_prefetch(row, 0, 1);              // global_prefetch_b8: warm gather row

    float cb0 = cbfT[s0 * K + k];               // broadcast-ish, cached
    float cb1 = cbfT[s1 * K + k];

    v2f rv = *(const v2f*)(row + j);            // L2-resident gather, 8B-aligned
    v2f ov;
    ov.x = rv.x * cb0;
    ov.y = rv.y * cb1;
    __builtin_nontemporal_store(ov, (v2f*)(out + (size_t)k * SR + j)); // TH=NT b64 store
}

// --- fallback: fused scalar path if ws can't hold the tables ---
__global__ void sbl_fused_kernel(const float* __restrict__ dist,
                                 const float* __restrict__ angles,
                                 const long long* __restrict__ aidx,
                                 const float* __restrict__ normt,
                                 float* __restrict__ out,
                                 unsigned int K, unsigned int total /* = K*42 */) {
    unsigned int gid = blockIdx.x * blockDim.x + threadIdx.x;
    if (gid >= total) return;
    unsigned int k = gid / SR;
    unsigned int j = gid - k * SR;
    unsigned int s = j / NUM_R;

    long long idx = aidx[(size_t)K + k];
    float u = dist[idx] * INV_CUTOFF;
    float rv = envelope(u) * normt[j] * sph_jn<NUM_S - 1>((int)s, u * c_zeros[j]);

    float x = __cosf(angles[k]);
    float pm1 = 1.0f, pc = x;
    float p = (s == 0) ? pm1 : pc;
#pragma unroll
    for (int l = 2; l < NUM_S; ++l) {
        float pl = ((float)(2 * l - 1) * x * pc - (float)(l - 1) * pm1) * (1.0f / (float)l);
        pm1 = pc; pc = pl;
        p = (s == (unsigned)l) ? pl : p;
    }
    __builtin_nontemporal_store(rv * p * c_coef[s], &out[gid]);
}

extern "C" void kernel_launch(void* const* d_in, const int* in_sizes, int n_in,
                              void* d_out, int out_size, void* d_ws, size_t ws_size,
                              hipStream_t stream) {
    const float*     dist   = (const float*)d_in[0];      // [M,1] f32
    const float*     angles = (const float*)d_in[1];      // [K,1] f32
    const long long* aidx   = (const long long*)d_in[2];  // [2,K] i64
    float*           out    = (float*)d_out;              // [K,42] f32

    const unsigned int M = (unsigned int)in_sizes[0];
    const unsigned int K = (unsigned int)in_sizes[1];

    // ws layout: [norm: 42 f32][pad to 256][cbf_T: 7*K f32][pad to 256][rbf: M*42 f32]
    float* normt = (float*)d_ws;
    size_t off_cbf = 256;
    size_t cbf_bytes = ((size_t)NUM_S * K * sizeof(float) + 255) & ~(size_t)255;
    size_t off_rbf = off_cbf + cbf_bytes;
    float* cbfT = (float*)((char*)d_ws + off_cbf);
    float* rbf  = (float*)((char*)d_ws + off_rbf);
    const size_t need = off_rbf + (size_t)M * SR * sizeof(float);

    sbl_init_norm<<<1, 64, 0, stream>>>(normt);

    if (ws_size >= need) {
        const unsigned int totalB = M * SR2;
        sbl_rbf_kernel<<<(totalB + 255u) / 256u, 256, 0, stream>>>(dist, normt, rbf, totalB);
        sbl_cbf_kernel<<<(K + 255u) / 256u, 256, 0, stream>>>(angles, cbfT, K);
        const unsigned int totalD = K * SR2;
        sbl_out_kernel<<<(totalD + 255u) / 256u, 256, 0, stream>>>(aidx, rbf, cbfT, out, K, totalD);
    } else {
        const unsigned int totalF = K * SR;
        sbl_fused_kernel<<<(totalF + 255u) / 256u, 256, 0, stream>>>(dist, angles, aidx, normt, out, K, totalF);
    }
}